// Conv3dFFT_88622355186034
// MI455X (gfx1250) — compile-verified
//
#include <hip/hip_runtime.h>

// ============================================================================
// FFT-based 3D circular convolution for MI455X (gfx1250).
//
//   out[oc,ic] = stride2(slice(irfftn(rfftn(pad(x[ic])) * rfftn(w[oc,ic]))))
//
// 64-point DFT passes are 64x64 matmuls on V_WMMA_F32_16X16X4_F32 (wave32).
// Each wave owns a 64(freq) x 16(lines) complex tile.  Twiddles come from a
// 64-entry (cos,sin) LDS table (ds_load_b64) -- only 64 distinct angles exist
// since theta = 2*pi*((pos*k) mod 64)/64.  Memory-bound by design
// (~3 GB traffic @ 23.3 TB/s  ->  ~130 us floor).
//
// Workspace requirement: 4*32*139392 + 4*128*139392 floats  (~357 MB).
// ============================================================================

typedef float v2f __attribute__((ext_vector_type(2)));
typedef float v8f __attribute__((ext_vector_type(8)));

#define TWOPI_64 0.09817477042468103f   // 2*pi/64

// D = A(16x4) * B(4x16) + C  on the CDNA5 matrix pipe.
__device__ __forceinline__ v8f wmma4(v2f a, v2f b, v8f c) {
  // 8-arg VOP3P form: (neg_a, A, neg_b, B, c_mod, C, reuse_a, reuse_b)
  return __builtin_amdgcn_wmma_f32_16x16x4_f32(false, a, false, b, (short)0, c,
                                               false, false);
}

// Fill the 64-entry twiddle table (cos, sin) and sync.  Must run before any
// wave-uniform early exit so the workgroup barrier is well formed.
__device__ __forceinline__ void fill_twiddles(float2* s_tw) {
  const int tid = threadIdx.x;
  if (tid < 64) {
    float sv, cv;
    __sincosf(TWOPI_64 * (float)tid, &sv, &cv);
    s_tw[tid] = make_float2(cv, sv);
  }
  __syncthreads();
}

// ----------------------------------------------------------------------------
// Pass 1: real 64-point rFFT along contiguous z axis.  MT=3 freq tiles
// (rows 0..47 computed, rows 0..32 stored).  PADDED=true reads the 62^3 input
// embedded at offset +1 inside a logical 64^3 zero-padded volume.
// ----------------------------------------------------------------------------
template <bool PADDED>
__global__ void rfft_z_kernel(const float* __restrict__ in,
                              float* __restrict__ out_re,
                              float* __restrict__ out_im, int n_lines) {
  __shared__ float2 s_tw[64];
  fill_twiddles(s_tw);

  const int lane = threadIdx.x & 31;
  const int wave = threadIdx.x >> 5;
  const int lh = lane >> 4;        // K-half selector for A/B fragments
  const int li = lane & 15;        // M row (A) / N column (B)
  const int b0 = (blockIdx.x * 8 + wave) * 16;
  if (b0 >= n_lines) return;       // wave-uniform exit: EXEC stays all-ones
  const int bcol = b0 + li;
  const bool bval = bcol < n_lines;
  const int bc = bval ? bcol : (n_lines - 1);

  long long ibase;
  bool linevalid = true;
  if (PADDED) {
    const int ic = bc >> 12;            // (32,64,64) line enumeration
    const int xp = (bc >> 6) & 63;
    const int yp = bc & 63;
    linevalid = (xp >= 1) && (xp <= 62) && (yp >= 1) && (yp <= 62);
    ibase = (long long)ic * 238328 + (long long)(xp - 1) * 3844 +
            (long long)(yp - 1) * 62;   // x is (32,62,62,62)
  } else {
    ibase = (long long)bc * 64;         // weight lines are contiguous
  }

  v8f aR[3] = {};
  v8f aI[3] = {};

  for (int kk = 0; kk < 16; ++kk) {     // K = 64 in steps of 4
    const int kb = kk * 4 + lh * 2;
    v2f Br;
    if (PADDED) {
      const bool v0 = linevalid && (kb >= 1) && (kb <= 62);
      const bool v1 = linevalid && (kb + 1 >= 1) && (kb + 1 <= 62);
      Br.x = v0 ? in[ibase + (kb - 1)] : 0.f;
      Br.y = v1 ? in[ibase + kb] : 0.f;
    } else {
      Br.x = in[ibase + kb];
      Br.y = in[ibase + kb + 1];
    }
#pragma unroll
    for (int t = 0; t < 3; ++t) {
      const int row = t * 16 + li;      // output frequency
      const float2 w0 = s_tw[(row * kb) & 63];
      const float2 w1 = s_tw[(row * (kb + 1)) & 63];
      v2f Ar; Ar.x = w0.x;  Ar.y = w1.x;    // Re D = cos
      v2f Ai; Ai.x = -w0.y; Ai.y = -w1.y;   // Im D = -sin (forward)
      aR[t] = wmma4(Ar, Br, aR[t]);
      aI[t] = wmma4(Ai, Br, aI[t]);
    }
  }

  if (bval) {
    const long long ob = (long long)bc * 33;
#pragma unroll
    for (int t = 0; t < 3; ++t)
#pragma unroll
      for (int j = 0; j < 8; ++j) {
        const int f = t * 16 + j + lh * 8;
        if (f < 33) {
          out_re[ob + f] = aR[t][j];
          out_im[ob + f] = aI[t][j];
        }
      }
  }
}

// ----------------------------------------------------------------------------
// Generic batched complex 64-point DFT along a strided axis.
//   line b: base = (b/d1)*s0 + ((b%d1)/d2)*s1 + (b%d2)*s2, element stride es.
//   output: same decomposition with t0/t1/t2, row r at base + r*oes.
// INVERSE: e^{+i}, output rows restricted to strided positions pos = 1+2*m.
// FMUL: after the DFT, multiply by XF[ic = (b/d1)%32] (fused frequency-domain
//       product, hardcoded XF strides 135168 / 2112 / 1).
// ----------------------------------------------------------------------------
template <int MT, bool INVERSE, bool FMUL>
__global__ void cdft_pass_kernel(
    const float* __restrict__ in_re, const float* __restrict__ in_im,
    float* __restrict__ out_re, float* __restrict__ out_im,
    const float* __restrict__ mul_re, const float* __restrict__ mul_im,
    int n_lines, int d1, int d2, long long s0, long long s1, long long s2,
    long long es, long long t0, long long t1, long long t2, long long oes,
    int store_rows) {
  __shared__ float2 s_tw[64];
  fill_twiddles(s_tw);

  const int lane = threadIdx.x & 31;
  const int wave = threadIdx.x >> 5;
  const int lh = lane >> 4;
  const int li = lane & 15;
  const int b0 = (blockIdx.x * 8 + wave) * 16;
  if (b0 >= n_lines) return;
  const int bcol = b0 + li;
  const bool bval = bcol < n_lines;
  const int bc = bval ? bcol : (n_lines - 1);

  const int q1 = bc / d1, r1 = bc % d1;
  const int q2 = r1 / d2, r2 = r1 % d2;
  const long long ibase =
      (long long)q1 * s0 + (long long)q2 * s1 + (long long)r2 * s2;
  const long long obase =
      (long long)q1 * t0 + (long long)q2 * t1 + (long long)r2 * t2;

  v8f aR[MT] = {};
  v8f aI[MT] = {};

  for (int kk = 0; kk < 16; ++kk) {
    const int kb = kk * 4 + lh * 2;
    const long long a0 = ibase + (long long)kb * es;
    // Speculative prefetch two K-steps ahead (lines stride up to 8.4 KB, so
    // the hardware cannot guess the stream) -> global_prefetch_b8.
    if (kk < 14) {
      __builtin_prefetch(in_re + a0 + 8 * es, 0, 1);
      __builtin_prefetch(in_im + a0 + 8 * es, 0, 1);
    }
    v2f Br, Bi;
    Br.x = in_re[a0];      Br.y = in_re[a0 + es];
    Bi.x = in_im[a0];      Bi.y = in_im[a0 + es];
#pragma unroll
    for (int t = 0; t < MT; ++t) {
      const int row = t * 16 + li;
      const int pos = INVERSE ? (1 + 2 * row) : row;  // strided output rows
      const float2 w0 = s_tw[(pos * kb) & 63];
      const float2 w1 = s_tw[(pos * (kb + 1)) & 63];
      v2f Ar;  Ar.x = w0.x;  Ar.y = w1.x;
      v2f Ai, nAi;
      if (INVERSE) { Ai.x = w0.y;  Ai.y = w1.y;  nAi.x = -w0.y; nAi.y = -w1.y; }
      else         { Ai.x = -w0.y; Ai.y = -w1.y; nAi.x = w0.y;  nAi.y = w1.y; }
      // complex GEMM: Cr += Ar*Br - Ai*Bi ; Ci += Ar*Bi + Ai*Br
      aR[t] = wmma4(Ar, Br, aR[t]);
      aR[t] = wmma4(nAi, Bi, aR[t]);
      aI[t] = wmma4(Ar, Bi, aI[t]);
      aI[t] = wmma4(Ai, Br, aI[t]);
    }
  }

  if (FMUL) {  // fused (oc,ic)-spectrum x XF[ic] product
    const long long mbase = (long long)(q1 & 31) * 135168LL + (long long)r1;
#pragma unroll
    for (int t = 0; t < MT; ++t)
#pragma unroll
      for (int j = 0; j < 8; ++j) {
        const int row = t * 16 + j + lh * 8;
        const long long ma = mbase + (long long)row * 2112LL;
        const float xr = mul_re[ma], xi = mul_im[ma];
        const float wr = aR[t][j], wi = aI[t][j];
        aR[t][j] = wr * xr - wi * xi;
        aI[t][j] = wr * xi + wi * xr;
      }
  }

  if (bval) {
#pragma unroll
    for (int t = 0; t < MT; ++t)
#pragma unroll
      for (int j = 0; j < 8; ++j) {
        const int row = t * 16 + j + lh * 8;
        if (row < store_rows) {
          out_re[obase + (long long)row * oes] = aR[t][j];
          out_im[obase + (long long)row * oes] = aI[t][j];
        }
      }
  }
}

// ----------------------------------------------------------------------------
// Final pass: Hermitian 33->64 inverse rFFT along z, fused with the 1/64^3
// scale, the [1:-1:2] z selection (pos = 1+2m), and the store to d_out.
//   out[pos] = S * sum_f w_f * (Xr_f cos(2pi f pos/64) - Xi_f sin(...)),
//   w_f = 1 for f in {0,32}, 2 otherwise.
// ----------------------------------------------------------------------------
__global__ void irfft_z_kernel(const float* __restrict__ in_re,
                               const float* __restrict__ in_im,
                               float* __restrict__ out, int n_lines, int d1,
                               int d2, long long s0, long long s1, long long s2,
                               long long t0, long long t1, long long t2) {
  __shared__ float2 s_tw[64];
  fill_twiddles(s_tw);

  const int lane = threadIdx.x & 31;
  const int wave = threadIdx.x >> 5;
  const int lh = lane >> 4;
  const int li = lane & 15;
  const int b0 = (blockIdx.x * 8 + wave) * 16;
  if (b0 >= n_lines) return;
  const int bcol = b0 + li;
  const bool bval = bcol < n_lines;
  const int bc = bval ? bcol : (n_lines - 1);

  const int q1 = bc / d1, r1 = bc % d1;
  const int q2 = r1 / d2, r2 = r1 % d2;
  const long long ibase =
      (long long)q1 * s0 + (long long)q2 * s1 + (long long)r2 * s2;  // f str 1
  const long long obase =
      (long long)q1 * t0 + (long long)q2 * t1 + (long long)r2 * t2;  // z str 1

  v8f acc[2] = {};
  const float S = 3.814697265625e-06f;  // 1 / 64^3

  for (int kk = 0; kk < 9; ++kk) {      // K = 33, padded to 36
    const int k0 = kk * 4 + lh * 2;
    const int k1 = k0 + 1;
    const int k0c = k0 > 32 ? 32 : k0;  // clamp addresses; A column is 0 there
    const int k1c = k1 > 32 ? 32 : k1;
    v2f Br, Bi;
    Br.x = in_re[ibase + k0c];  Br.y = in_re[ibase + k1c];
    Bi.x = in_im[ibase + k0c];  Bi.y = in_im[ibase + k1c];
    const float w0 = (k0 == 0 || k0 == 32) ? S : (k0 < 33 ? 2.f * S : 0.f);
    const float w1 = (k1 == 0 || k1 == 32) ? S : (k1 < 33 ? 2.f * S : 0.f);
#pragma unroll
    for (int t = 0; t < 2; ++t) {
      const int pos = 1 + 2 * (t * 16 + li);  // strided z output position
      const float2 c0 = s_tw[(pos * k0) & 63];
      const float2 c1 = s_tw[(pos * k1) & 63];
      v2f Ar; Ar.x = w0 * c0.x;    Ar.y = w1 * c1.x;
      v2f As; As.x = -(w0 * c0.y); As.y = -(w1 * c1.y);
      acc[t] = wmma4(Ar, Br, acc[t]);
      acc[t] = wmma4(As, Bi, acc[t]);
    }
  }

  if (bval) {
#pragma unroll
    for (int t = 0; t < 2; ++t)
#pragma unroll
      for (int j = 0; j < 8; ++j) {
        const int m = t * 16 + j + lh * 8;
        if (m < 31) out[obase + m] = acc[t][j];
      }
  }
}

// ============================================================================
// Orchestration.  Workspace layout (floats):
//   XA/XB re+im : 4 * 32*139392   (x-spectrum ping-pong; XA ends as XF)
//   WA/WB re+im : 4 * 128*139392  (weight chunk ping-pong, 4 chunks of 128)
// ============================================================================
extern "C" void kernel_launch(void* const* d_in, const int* in_sizes, int n_in,
                              void* d_out, int out_size, void* d_ws,
                              size_t ws_size, hipStream_t stream) {
  const float* x = (const float*)d_in[0];    // (1,32,62,62,62)
  const float* w = (const float*)d_in[1];    // (16,32,64,64,64)
  float* out = (float*)d_out;                // (16,32,31,31,31)

  float* ws = (float*)d_ws;
  const long long NXp = 32LL * 139392;       // 64*64*33 per volume
  const long long NWp = 128LL * 139392;
  float* XA_re = ws;            float* XA_im = XA_re + NXp;
  float* XB_re = XA_im + NXp;   float* XB_im = XB_re + NXp;
  float* WA_re = XB_im + NXp;   float* WA_im = WA_re + NWp;
  float* WB_re = WA_im + NWp;   float* WB_im = WB_re + NWp;

  const dim3 blk(256);  // 8 wave32 waves; each wave owns 16 lines

  // ---- x pipeline: pad + rfft(z), DFT(y), DFT(x)  ->  XF in XA ----
  rfft_z_kernel<true><<<1024, blk, 0, stream>>>(x, XA_re, XA_im, 131072);
  cdft_pass_kernel<4, false, false><<<528, blk, 0, stream>>>(
      XA_re, XA_im, XB_re, XB_im, nullptr, nullptr, 67584, 2112, 33, 135168,
      2112, 1, 33, 135168, 2112, 1, 33, 64);
  cdft_pass_kernel<4, false, false><<<528, blk, 0, stream>>>(
      XB_re, XB_im, XA_re, XA_im, nullptr, nullptr, 67584, 2112, 33, 135168,
      33, 1, 2112, 135168, 33, 1, 2112, 64);

  // ---- weight pipeline, 4 chunks of 128 (oc,ic) volumes ----
  for (int c = 0; c < 4; ++c) {
    const float* wc = w + (long long)c * 128 * 262144;
    float* oc = out + (long long)c * 128 * 29791;

    rfft_z_kernel<false><<<4096, blk, 0, stream>>>(wc, WA_re, WA_im, 524288);
    cdft_pass_kernel<4, false, false><<<2112, blk, 0, stream>>>(
        WA_re, WA_im, WB_re, WB_im, nullptr, nullptr, 270336, 2112, 33, 135168,
        2112, 1, 33, 135168, 2112, 1, 33, 64);
    // forward DFT(x) fused with pointwise multiply by XF[ic]
    cdft_pass_kernel<4, false, true><<<2112, blk, 0, stream>>>(
        WB_re, WB_im, WA_re, WA_im, XA_re, XA_im, 270336, 2112, 33, 135168, 33,
        1, 2112, 135168, 33, 1, 2112, 64);
    // inverse DFT(x), only the 31 strided spatial positions
    cdft_pass_kernel<2, true, false><<<2112, blk, 0, stream>>>(
        WA_re, WA_im, WB_re, WB_im, nullptr, nullptr, 270336, 2112, 33, 135168,
        33, 1, 2112, 65472, 33, 1, 2112, 31);
    // inverse DFT(y), only the 31 strided spatial positions
    cdft_pass_kernel<2, true, false><<<992, blk, 0, stream>>>(
        WB_re, WB_im, WA_re, WA_im, nullptr, nullptr, 126864, 1023, 33, 65472,
        2112, 1, 33, 31713, 1023, 1, 33, 31);
    // inverse rfft(z) fused with scale + stride-2 selection -> d_out
    irfft_z_kernel<<<961, blk, 0, stream>>>(WA_re, WA_im, oc, 123008, 961, 31,
                                            31713, 1023, 33, 29791, 961, 31);
  }
}